// AnomalyMapGenerator_44461501448652
// MI455X (gfx1250) — compile-verified
//
#include <hip/hip_runtime.h>

typedef float v2f __attribute__((ext_vector_type(2)));
typedef float v8f __attribute__((ext_vector_type(8)));

namespace {
constexpr int kB  = 32;    // batch
constexpr int kC  = 200;   // channels
constexpr int kCP = 210;   // padded channel stride in LDS (bank-spreading, even)
constexpr int kHW = 3136;  // 56*56 pixels
constexpr int kNW = 13;    // waves per block = ceil(208/16) d-tiles
constexpr int kT  = kNW * 32;   // 416 threads
}

__global__ __launch_bounds__(kT)
void padim_mahalanobis_wmma_f32(const float* __restrict__ emb,
                                const float* __restrict__ mean,
                                const float* __restrict__ invcov,
                                float* __restrict__ out)
{
  __shared__ float sDelta[kB * kCP];   // 32 x 210 f32 (~26.9 KB); cols >=200 stay zero
  __shared__ float sPart[kNW * kB];    // per-wave partial distances

  const int p   = blockIdx.x;          // pixel
  const int tid = threadIdx.x;

  // ---- Phase 1: delta = emb[:, :, p] - mean[:, p] into LDS, zero-padded to 210 cols
  for (int idx = tid; idx < kB * kCP; idx += kT) {
    const int b = idx / kCP;
    const int c = idx - b * kCP;
    float v = 0.0f;
    if (c < kC)
      v = emb[((size_t)b * kC + c) * kHW + p] - mean[(size_t)c * kHW + p];
    sDelta[idx] = v;
  }
  __syncthreads();

  // ---- Phase 2: per-wave GEMM tile  M(b,d) = sum_c delta[b,c] * A_p[c,d]
  const int wave = tid >> 5;           // d-tile index 0..12
  const int lane = tid & 31;
  const int l15  = lane & 15;
  const int hi   = lane >> 4;          // 0: lanes 0-15 (K=0,1) ; 1: lanes 16-31 (K=2,3)
  const int kk   = hi << 1;

  const int d0   = wave * 16;
  const int dcol = d0 + l15;                       // this lane's output column
  const int dcl  = dcol < kC ? dcol : (kC - 1);    // clamp: padded cols killed in epilogue

  const float* Ap    = invcov + (size_t)p * (kC * kC);
  const float* bptr  = Ap + kk * kC + dcl;               // rows kk,kk+1 ; col dcl
  const float* drow0 = &sDelta[l15 * kCP + kk];          // A-operand rows b = 0..15
  const float* drow1 = &sDelta[(l15 + 16) * kCP + kk];   // A-operand rows b = 16..31

  v8f acc0 = {0.f,0.f,0.f,0.f,0.f,0.f,0.f,0.f};
  v8f acc1 = {0.f,0.f,0.f,0.f,0.f,0.f,0.f,0.f};

#pragma unroll 5
  for (int c0 = 0; c0 < kC; c0 += 4) {
    // A operands (delta): lane layout = 16x4 f32 A-matrix (K split 0,1 / 2,3 by half-wave)
    const v2f a0 = *(const v2f*)(drow0 + c0);      // ds_load_b64
    const v2f a1 = *(const v2f*)(drow1 + c0);
    // B operand (inv covariance): rows c0+kk, c0+kk+1 at column dcol (coalesced rows)
    v2f bb;
    bb.x = bptr[c0 * kC];
    bb.y = bptr[c0 * kC + kC];
    acc0 = __builtin_amdgcn_wmma_f32_16x16x4_f32(false, a0, false, bb, (short)0, acc0, false, false);
    acc1 = __builtin_amdgcn_wmma_f32_16x16x4_f32(false, a1, false, bb, (short)0, acc1, false, false);
  }

  // ---- Phase 3: dist_b += sum_d M[b,d] * delta[b,d] ; reduce across 16-lane halves
  float red0[8], red1[8];
#pragma unroll
  for (int r = 0; r < 8; ++r) {
    const int b0 = r + (hi << 3);          // C/D layout: VGPR r -> M = r (+8 for upper half)
    const int b1 = 16 + b0;
    float v0 = acc0[r] * sDelta[b0 * kCP + dcol];   // zero for padded d
    float v1 = acc1[r] * sDelta[b1 * kCP + dcol];
#pragma unroll
    for (int m = 1; m <= 8; m <<= 1) {     // butterfly within each 16-lane half
      v0 += __shfl_xor(v0, m, 32);
      v1 += __shfl_xor(v1, m, 32);
    }
    red0[r] = v0;
    red1[r] = v1;
  }
  if (l15 == 0) {                          // lanes 0 and 16 publish their half's rows
#pragma unroll
    for (int r = 0; r < 8; ++r) {
      sPart[wave * kB + (hi << 3) + r]      = red0[r];
      sPart[wave * kB + 16 + (hi << 3) + r] = red1[r];
    }
  }
  __syncthreads();

  // ---- Phase 4: sum the 13 wave partials, sqrt(clip), write (B,1,H,W)
  if (tid < kB) {
    float s = 0.0f;
#pragma unroll
    for (int w = 0; w < kNW; ++w) s += sPart[w * kB + tid];
    out[(size_t)tid * kHW + p] = sqrtf(fmaxf(s, 0.0f));
  }
}

extern "C" void kernel_launch(void* const* d_in, const int* in_sizes, int n_in,
                              void* d_out, int out_size, void* d_ws, size_t ws_size,
                              hipStream_t stream) {
  (void)in_sizes; (void)n_in; (void)d_ws; (void)ws_size; (void)out_size;
  const float* emb    = (const float*)d_in[0];
  const float* mean   = (const float*)d_in[1];
  const float* invcov = (const float*)d_in[2];
  float* out          = (float*)d_out;
  hipLaunchKernelGGL(padim_mahalanobis_wmma_f32, dim3(kHW), dim3(kT), 0, stream,
                     emb, mean, invcov, out);
}